// GeoGraphPNet_31207232373349
// MI455X (gfx1250) — compile-verified
//
#include <hip/hip_runtime.h>
#include <math.h>

// ---------------- problem constants ----------------
constexpr int  N0v = 1638400, N1v = 163840, N2v = 16384;
constexpr long long E0v = 1638400LL;   // N1*DEG
constexpr long long E1v = 163840LL;    // N2*DEG
constexpr float EPSv = 1e-5f;

// ---------------- workspace layout (float offsets) ----------------
constexpr size_t OFF_NUM0 = 0;                                   // N1*32
constexpr size_t OFF_DEN0 = OFF_NUM0 + (size_t)N1v * 32;         // N1
constexpr size_t OFF_H    = OFF_DEN0 + (size_t)N1v;              // N1*64
constexpr size_t OFF_NUM1 = OFF_H    + (size_t)N1v * 64;         // N2*64
constexpr size_t OFF_DEN1 = OFF_NUM1 + (size_t)N2v * 64;         // N2
constexpr size_t OFF_XIN  = OFF_DEN1 + (size_t)N2v;              // N2*48
constexpr size_t OFF_ATTWP= OFF_XIN  + (size_t)N2v * 48;         // 48*48
constexpr size_t OFF_W0P  = OFF_ATTWP + 2304;                    // 48*128
constexpr size_t OFF_WL2P = OFF_W0P   + 6144;                    // 128*48
constexpr size_t OFF_ATTBP= OFF_WL2P  + 6144;                    // 48
constexpr size_t OFF_CL2P = OFF_ATTBP + 48;                      // 48
constexpr size_t OFF_STATS= OFF_CL2P  + 48;                      // 7 layers * 256
// aliased regions (lifetimes do not overlap with their hosts):
constexpr size_t OFF_YATT = OFF_NUM0;                            // N2*48 (num0 dead)
constexpr size_t OFF_Z    = OFF_H;                               // N2*128 (h dead after gcn1)
constexpr size_t OFF_H0   = OFF_Z  + (size_t)N2v * 128;
constexpr size_t OFF_H1   = OFF_H0 + (size_t)N2v * 128;
constexpr size_t OFF_H2   = OFF_H1 + (size_t)N2v * 64;
constexpr size_t OFF_H3   = OFF_H2 + (size_t)N2v * 32;
constexpr size_t OFF_H4   = OFF_H3 + (size_t)N2v * 64;
constexpr size_t OFF_H5   = OFF_H4 + (size_t)N2v * 128;          // N2*48, fits in h region

typedef __attribute__((ext_vector_type(2))) float v2f;
typedef __attribute__((ext_vector_type(8))) float v8f;

#define WMMA_F32X4(a, b, c) \
  __builtin_amdgcn_wmma_f32_16x16x4_f32(false, (a), false, (b), (short)0, (c), false, false)

// ---------------- utility kernels ----------------
__global__ void k_zero(float* __restrict__ p, long long n) {
  long long i = (long long)blockIdx.x * blockDim.x + threadIdx.x;
  long long st = (long long)gridDim.x * blockDim.x;
  for (; i < n; i += st) p[i] = 0.0f;
}

// pad attW(33x33)->48x48, W0(33x128)->48x128, Wlast2(128x33)->128x48, attb/clast2 -> 48
__global__ void k_prep_params(const float* __restrict__ attW, const float* __restrict__ attb,
                              const float* __restrict__ W0,   const float* __restrict__ Wlast2,
                              const float* __restrict__ clast2,
                              float* __restrict__ attWp, float* __restrict__ W0p,
                              float* __restrict__ Wl2p,  float* __restrict__ attbp,
                              float* __restrict__ cl2p) {
  int i = blockIdx.x * 256 + threadIdx.x;
  if (i < 2304) { int r = i / 48, c = i % 48;
    attWp[i] = (r < 33 && c < 33) ? attW[r * 33 + c] : 0.0f; return; }
  i -= 2304;
  if (i < 6144) { int r = i / 128;
    W0p[i + 0] = (r < 33) ? W0[i] : 0.0f; return; }
  i -= 6144;
  if (i < 6144) { int r = i / 48, c = i % 48;
    Wl2p[i] = (c < 33) ? Wlast2[r * 33 + c] : 0.0f; return; }
  i -= 6144;
  if (i < 48) { attbp[i] = (i < 33) ? attb[i] : 0.0f; return; }
  i -= 48;
  if (i < 48) { cl2p[i] = (i < 33) ? clast2[i] : 0.0f; return; }
}

// ---------------- edge scatter (segment sums) ----------------
__global__ void k_scatter0(const float* __restrict__ x, const int* __restrict__ src,
                           const int* __restrict__ dst, const float* __restrict__ w,
                           float* __restrict__ num, float* __restrict__ den, long long nwork) {
  long long i = (long long)blockIdx.x * blockDim.x + threadIdx.x;
  long long st = (long long)gridDim.x * blockDim.x;
  for (; i < nwork; i += st) {
    long long e = i >> 5; int c = (int)(i & 31);
    int s = src[e], d = dst[e];
    float wv = w[e];
    atomicAdd(&num[(size_t)d * 32 + c], wv * x[(size_t)s * 32 + c]);
    if (c == 0) atomicAdd(&den[d], wv);
  }
}

__global__ void k_scatter1(const float* __restrict__ h, const int* __restrict__ src,
                           const int* __restrict__ dst, const float* __restrict__ w,
                           float* __restrict__ num, float* __restrict__ den, long long nwork) {
  long long i = (long long)blockIdx.x * blockDim.x + threadIdx.x;
  long long st = (long long)gridDim.x * blockDim.x;
  for (; i < nwork; i += st) {
    long long e = i >> 6; int c = (int)(i & 63);
    int s = src[e], d = dst[e];
    float wv = w[e];
    atomicAdd(&num[(size_t)d * 64 + c], wv * h[(size_t)s * 64 + c]);
    if (c == 0) atomicAdd(&den[d], wv);
  }
}

// ---------------- GCN layer 0 dense: h = relu([num/den | x] @ [[Wn0];[Wr0]] + b0) ----------------
// grid (N1/64, 4), block (32,4). One wave = one 16x16 output tile. f32 WMMA, K=64.
__global__ void k_gcn0(const float* __restrict__ num0, const float* __restrict__ den0,
                       const float* __restrict__ x,
                       const float* __restrict__ Wn0, const float* __restrict__ Wr0,
                       const float* __restrict__ b0, float* __restrict__ h) {
  const int lane = threadIdx.x;
  const int half = lane >> 4, l15 = lane & 15;
  const int m0 = (blockIdx.x * 4 + threadIdx.y) * 16;
  const int n0 = blockIdx.y * 16;
  const int row = m0 + l15;
  const int col = n0 + l15;
  const float rden = 1.0f / fmaxf(den0[row], 1e-12f);
  const float* arow0 = num0 + (size_t)row * 32 + 2 * half;
  const float* arow1 = x    + (size_t)row * 32 + 2 * half;
  v8f acc = {};
  #pragma unroll
  for (int k = 0; k < 32; k += 4) {                 // agg @ Wn0
    v2f a, b;
    a.x = arow0[k] * rden; a.y = arow0[k + 1] * rden;
    const float* wp = Wn0 + (size_t)(k + 2 * half) * 64 + col;
    b.x = wp[0]; b.y = wp[64];
    acc = WMMA_F32X4(a, b, acc);
  }
  #pragma unroll
  for (int k = 0; k < 32; k += 4) {                 // x_tgt @ Wr0
    v2f a, b;
    a.x = arow1[k]; a.y = arow1[k + 1];
    const float* wp = Wr0 + (size_t)(k + 2 * half) * 64 + col;
    b.x = wp[0]; b.y = wp[64];
    acc = WMMA_F32X4(a, b, acc);
  }
  const float bv = b0[col];
  #pragma unroll
  for (int r = 0; r < 8; ++r) {
    float v = fmaxf(acc[r] + bv, 0.0f);
    h[(size_t)(m0 + r + 8 * half) * 64 + col] = v;
  }
}

// ---------------- GCN layer 1 + concat into xin (wave-per-row, K=128 -> scalar) ----------------
__global__ void k_gcn1_xin(const float* __restrict__ num1, const float* __restrict__ den1,
                           const float* __restrict__ h,
                           const float* __restrict__ Wn1, const float* __restrict__ Wr1,
                           const float* __restrict__ b1, const float* __restrict__ xnode,
                           float* __restrict__ xin) {
  const int lane = threadIdx.x & 31;
  const int wid  = threadIdx.x >> 5;
  const int row  = blockIdx.x * (blockDim.x >> 5) + wid;
  const float rden = 1.0f / fmaxf(den1[row], 1e-12f);
  float acc = 0.0f;
  for (int k = lane; k < 64; k += 32)
    acc += num1[(size_t)row * 64 + k] * rden * Wn1[k] + h[(size_t)row * 64 + k] * Wr1[k];
  for (int off = 16; off > 0; off >>= 1) acc += __shfl_xor(acc, off, 32);
  float gcn = acc + b1[0];
  xin[(size_t)row * 48 + lane] = xnode[(size_t)row * 32 + lane];
  if (lane == 0) xin[(size_t)row * 48 + 32] = gcn;
  if (lane >= 1 && lane < 16) xin[(size_t)row * 48 + 32 + lane] = 0.0f;
}

// ---------------- generic f32 WMMA GEMM: Z = act(X @ W + bias) ----------------
// grid (M/64, Nld/16), block (32,4)
__global__ void k_gemm(const float* __restrict__ X, int ldx,
                       const float* __restrict__ W, int ldw,
                       const float* __restrict__ bias,
                       float* __restrict__ Z, int ldz, int K, int relu) {
  const int lane = threadIdx.x;
  const int half = lane >> 4, l15 = lane & 15;
  const int m0 = (blockIdx.x * 4 + threadIdx.y) * 16;
  const int n0 = blockIdx.y * 16;
  const int col = n0 + l15;
  const float* xrow = X + (size_t)(m0 + l15) * ldx + 2 * half;
  v8f acc = {};
  for (int k = 0; k < K; k += 4) {
    v2f a, b;
    a.x = xrow[k]; a.y = xrow[k + 1];
    const float* wp = W + (size_t)(k + 2 * half) * ldw + col;
    b.x = wp[0]; b.y = wp[ldw];
    acc = WMMA_F32X4(a, b, acc);
  }
  const float bv = bias ? bias[col] : 0.0f;
  #pragma unroll
  for (int r = 0; r < 8; ++r) {
    float v = acc[r] + bv;
    if (relu) v = fmaxf(v, 0.0f);
    Z[(size_t)(m0 + r + 8 * half) * ldz + col] = v;
  }
}

// ---------------- attention softmax + residual + BN stats (thread per row) ----------------
__global__ void k_att_softmax(const float* __restrict__ xin, float* __restrict__ z,
                              float* __restrict__ sum, float* __restrict__ sumsq) {
  __shared__ float ls[48], lq[48];
  const int t = threadIdx.x;
  if (t < 48) { ls[t] = 0.0f; lq[t] = 0.0f; }
  __syncthreads();
  const int row = blockIdx.x * 256 + t;
  float v[33];
  float mx = -3.4e38f;
  for (int c = 0; c < 33; ++c) { v[c] = z[(size_t)row * 48 + c]; mx = fmaxf(mx, v[c]); }
  float s = 0.0f;
  for (int c = 0; c < 33; ++c) { float e = __expf(v[c] - mx); v[c] = e; s += e; }
  const float inv = 1.0f / s;
  for (int c = 0; c < 33; ++c) {
    float xv = xin[(size_t)row * 48 + c];
    float y = xv * (v[c] * inv) + xv;
    z[(size_t)row * 48 + c] = y;
    atomicAdd(&ls[c], y);
    atomicAdd(&lq[c], y * y);
  }
  for (int c = 33; c < 48; ++c) z[(size_t)row * 48 + c] = 0.0f;
  __syncthreads();
  if (t < 33) { atomicAdd(&sum[t], ls[t]); atomicAdd(&sumsq[t], lq[t]); }
}

// ---------------- per-column sums for batchnorm ----------------
__global__ void k_colstats(const float* __restrict__ Z, int ld, int n,
                           float* __restrict__ sum, float* __restrict__ sumsq) {
  __shared__ float ls[128], lq[128];
  const int t = threadIdx.x;
  for (int c = t; c < n; c += 256) { ls[c] = 0.0f; lq[c] = 0.0f; }
  __syncthreads();
  const int row = blockIdx.x * 256 + t;
  for (int c = 0; c < n; ++c) {
    float v = Z[(size_t)row * ld + c];
    atomicAdd(&ls[c], v);
    atomicAdd(&lq[c], v * v);
  }
  __syncthreads();
  for (int c = t; c < n; c += 256) { atomicAdd(&sum[c], ls[c]); atomicAdd(&sumsq[c], lq[c]); }
}

// ---------------- batchnorm apply (+ optional residual), zero pad columns ----------------
__global__ void k_bnapply(const float* __restrict__ Z, float* __restrict__ out,
                          int ld, int n, float invM,
                          const float* __restrict__ sum, const float* __restrict__ sumsq,
                          const float* __restrict__ g, const float* __restrict__ be,
                          const float* __restrict__ res, long long total) {
  long long i = (long long)blockIdx.x * blockDim.x + threadIdx.x;
  long long st = (long long)gridDim.x * blockDim.x;
  for (; i < total; i += st) {
    int c = (int)(i % ld);
    float v = 0.0f;
    if (c < n) {
      float m  = sum[c] * invM;
      float vr = sumsq[c] * invM - m * m;
      v = (Z[i] - m) * rsqrtf(vr + EPSv) * g[c] + be[c];
      if (res) v += res[i];
    }
    out[i] = v;
  }
}

// ---------------- final head: out = h5 @ Wlast + clast ; pdif = relu(pm25-pm10) ----------------
__global__ void k_final(const float* __restrict__ h5, const float* __restrict__ Wlast,
                        const float* __restrict__ clast, float* __restrict__ out) {
  const int r = blockIdx.x * 256 + threadIdx.x;
  float a = clast[0], b = clast[1];
  for (int k = 0; k < 33; ++k) {
    float v = h5[(size_t)r * 48 + k];
    a += v * Wlast[k * 2 + 0];
    b += v * Wlast[k * 2 + 1];
  }
  out[(size_t)r * 3 + 0] = a;
  out[(size_t)r * 3 + 1] = b;
  out[(size_t)r * 3 + 2] = fmaxf(a - b, 0.0f);
}

// ---------------- host launcher ----------------
static void run_mlp_layer(hipStream_t stream, float* ws,
                          const float* X, int ldx, int K,
                          const float* W, int ldw, const float* bias,
                          int ldo, int nlog, const float* g, const float* be,
                          const float* res, int statIdx, float* Zbuf, float* Obuf) {
  dim3 gg(N2v / 64, ldo / 16), gb(32, 4);
  k_gemm<<<gg, gb, 0, stream>>>(X, ldx, W, ldw, bias, Zbuf, ldo, K, 1);
  float* sum = ws + OFF_STATS + (size_t)statIdx * 256;
  k_colstats<<<N2v / 256, 256, 0, stream>>>(Zbuf, ldo, nlog, sum, sum + 128);
  long long tot = (long long)N2v * ldo;
  k_bnapply<<<(int)(tot / 256), 256, 0, stream>>>(Zbuf, Obuf, ldo, nlog, 1.0f / N2v,
                                                  sum, sum + 128, g, be, res, tot);
}

extern "C" void kernel_launch(void* const* d_in, const int* in_sizes, int n_in,
                              void* d_out, int out_size, void* d_ws, size_t ws_size,
                              hipStream_t stream) {
  (void)in_sizes; (void)n_in; (void)out_size; (void)ws_size;
  const float* x     = (const float*)d_in[0];
  const int*   src0  = (const int*)d_in[1];
  const int*   dst0  = (const int*)d_in[2];
  const float* w0    = (const float*)d_in[3];
  const int*   src1  = (const int*)d_in[4];
  const int*   dst1  = (const int*)d_in[5];
  const float* w1    = (const float*)d_in[6];
  const float* xnode = (const float*)d_in[7];
  const float* Wn0   = (const float*)d_in[8];
  const float* b0    = (const float*)d_in[9];
  const float* Wr0   = (const float*)d_in[10];
  const float* Wn1   = (const float*)d_in[11];
  const float* b1    = (const float*)d_in[12];
  const float* Wr1   = (const float*)d_in[13];
  const float* attW  = (const float*)d_in[14];
  const float* attb  = (const float*)d_in[15];
  const float* attg  = (const float*)d_in[16];
  const float* attbe = (const float*)d_in[17];
  const float *Wm[5], *cm[5], *gm[5], *bem[5];
  for (int i = 0; i < 5; ++i) {
    Wm[i]  = (const float*)d_in[18 + 4 * i];
    cm[i]  = (const float*)d_in[19 + 4 * i];
    gm[i]  = (const float*)d_in[20 + 4 * i];
    bem[i] = (const float*)d_in[21 + 4 * i];
  }
  const float* Wlast2 = (const float*)d_in[38];
  const float* clast2 = (const float*)d_in[39];
  const float* g5     = (const float*)d_in[40];
  const float* be5    = (const float*)d_in[41];
  const float* Wlast  = (const float*)d_in[42];
  const float* clast  = (const float*)d_in[43];

  float* ws  = (float*)d_ws;
  float* out = (float*)d_out;

  // zero accumulators + stats
  k_zero<<<2048, 256, 0, stream>>>(ws + OFF_NUM0, (long long)N1v * 33);  // num0 + den0 contiguous
  k_zero<<<512,  256, 0, stream>>>(ws + OFF_NUM1, (long long)N2v * 65);  // num1 + den1 contiguous
  k_zero<<<8,    256, 0, stream>>>(ws + OFF_STATS, 7 * 256);
  k_prep_params<<<(2304 + 6144 + 6144 + 48 + 48 + 255) / 256, 256, 0, stream>>>(
      attW, attb, Wm[0], Wlast2, clast2,
      ws + OFF_ATTWP, ws + OFF_W0P, ws + OFF_WL2P, ws + OFF_ATTBP, ws + OFF_CL2P);

  // GCN layer 0
  k_scatter0<<<4096, 256, 0, stream>>>(x, src0, dst0, w0, ws + OFF_NUM0, ws + OFF_DEN0, E0v * 32);
  k_gcn0<<<dim3(N1v / 64, 4), dim3(32, 4), 0, stream>>>(ws + OFF_NUM0, ws + OFF_DEN0, x,
                                                        Wn0, Wr0, b0, ws + OFF_H);
  // GCN layer 1 + concat
  k_scatter1<<<2048, 256, 0, stream>>>(ws + OFF_H, src1, dst1, w1, ws + OFF_NUM1, ws + OFF_DEN1,
                                       E1v * 64);
  k_gcn1_xin<<<N2v / 8, 256, 0, stream>>>(ws + OFF_NUM1, ws + OFF_DEN1, ws + OFF_H,
                                          Wn1, Wr1, b1, xnode, ws + OFF_XIN);

  // attention: logits -> softmax*x + x -> BN
  k_gemm<<<dim3(N2v / 64, 3), dim3(32, 4), 0, stream>>>(ws + OFF_XIN, 48, ws + OFF_ATTWP, 48,
                                                        ws + OFF_ATTBP, ws + OFF_Z, 48, 48, 0);
  {
    float* s0 = ws + OFF_STATS;
    k_att_softmax<<<N2v / 256, 256, 0, stream>>>(ws + OFF_XIN, ws + OFF_Z, s0, s0 + 128);
    long long tot = (long long)N2v * 48;
    k_bnapply<<<(int)(tot / 256), 256, 0, stream>>>(ws + OFF_Z, ws + OFF_YATT, 48, 33,
                                                    1.0f / N2v, s0, s0 + 128, attg, attbe,
                                                    nullptr, tot);
  }

  float* YATT = ws + OFF_YATT;
  float* Z    = ws + OFF_Z;
  float* H0   = ws + OFF_H0;
  float* H1   = ws + OFF_H1;
  float* H2   = ws + OFF_H2;
  float* H3   = ws + OFF_H3;
  float* H4   = ws + OFF_H4;
  float* H5   = ws + OFF_H5;

  // MLP with BN + residuals
  run_mlp_layer(stream, ws, YATT, 48, 48, ws + OFF_W0P, 128, cm[0], 128, 128, gm[0], bem[0],
                nullptr, 1, Z, H0);
  run_mlp_layer(stream, ws, H0, 128, 128, Wm[1], 64, cm[1], 64, 64, gm[1], bem[1],
                nullptr, 2, Z, H1);
  run_mlp_layer(stream, ws, H1, 64, 64, Wm[2], 32, cm[2], 32, 32, gm[2], bem[2],
                nullptr, 3, Z, H2);
  run_mlp_layer(stream, ws, H2, 32, 32, Wm[3], 64, cm[3], 64, 64, gm[3], bem[3],
                H1, 4, Z, H3);
  run_mlp_layer(stream, ws, H3, 64, 64, Wm[4], 128, cm[4], 128, 128, gm[4], bem[4],
                H0, 5, Z, H4);
  run_mlp_layer(stream, ws, H4, 128, 128, ws + OFF_WL2P, 48, ws + OFF_CL2P, 48, 33, g5, be5,
                YATT, 6, Z, H5);

  k_final<<<N2v / 256, 256, 0, stream>>>(H5, Wlast, clast, out);
}